// KVCacheManager_29025388986999
// MI455X (gfx1250) — compile-verified
//
#include <hip/hip_runtime.h>

// ---------------------------------------------------------------------------
// KV-cache accept/copy for MI455X (gfx1250).
// Pure bandwidth problem: ~1 GiB of traffic -> ~45us at 23.3 TB/s.
// Bulk copy uses the CDNA5 async global<->LDS DMA path (ASYNCcnt) with B128
// transfers; falls back to nontemporal B128 VGPR copies if the async builtins
// are unavailable in this toolchain.
// ---------------------------------------------------------------------------

typedef __attribute__((ext_vector_type(4))) unsigned int u32x4;
typedef __attribute__((ext_vector_type(4))) int          i32x4;

// Problem constants (from the reference).
#define L_DIM 8
#define R_DIM 32
#define T_DIM 512
#define H_DIM 8
#define D_DIM 128
#define ROW_ELEMS (H_DIM * D_DIM)        // 1024 fp16 per (l,r,t) row
#define ROW_VECS  (ROW_ELEMS / 8)        // 128 x 16B per row

#ifndef __has_builtin
#define __has_builtin(x) 0
#endif

#if __has_builtin(__builtin_amdgcn_global_load_async_to_lds_b128) && \
    __has_builtin(__builtin_amdgcn_global_store_async_from_lds_b128)
#define HAVE_ASYNC_LDS 1
#else
#define HAVE_ASYNC_LDS 0
#endif

// Builtin wants typed v4i32 pointers in the right address spaces.
// Chain casts: strip const -> generic i32x4* -> explicit addrspace cast.
#define GLOBAL_V4(p) ((__attribute__((address_space(1))) i32x4*)(i32x4*)(void*)(p))
#define LDS_V4(p)    ((__attribute__((address_space(3))) i32x4*)(i32x4*)(void*)(p))

__device__ __forceinline__ void wait_async0() {
#if __has_builtin(__builtin_amdgcn_s_wait_asynccnt)
  __builtin_amdgcn_s_wait_asynccnt(0);
#else
  asm volatile("s_wait_asynccnt 0x0" ::: "memory");
#endif
}

// ---------------------------------------------------------------------------
// Bulk copy: dst[0..n16) = src[0..n16) in 16-byte vectors.
// ---------------------------------------------------------------------------
#define CP_THREADS 256
#define CP_UNROLL  8
#define CP_TILE    (CP_THREADS * CP_UNROLL)

__global__ __launch_bounds__(CP_THREADS)
void bulk_copy_kernel(const u32x4* __restrict__ src, u32x4* __restrict__ dst,
                      long n16) {
#if HAVE_ASYNC_LDS
  __shared__ u32x4 lbuf[CP_TILE];  // 32 KB, lane-private slots, no barriers
#endif
  const int tid = threadIdx.x;
  long base = (long)blockIdx.x * CP_TILE;
  const long stride = (long)gridDim.x * CP_TILE;

  for (; base < n16; base += stride) {
    if (base + CP_TILE <= n16) {
#if HAVE_ASYNC_LDS
      // global -> LDS (async, B128 per lane)
      #pragma unroll
      for (int j = 0; j < CP_UNROLL; ++j) {
        long g = base + (long)j * CP_THREADS + tid;
        __builtin_amdgcn_global_load_async_to_lds_b128(
            GLOBAL_V4(src + g), LDS_V4(&lbuf[j * CP_THREADS + tid]), 0, 0);
      }
      wait_async0();
      // LDS -> global (async, B128 per lane)
      #pragma unroll
      for (int j = 0; j < CP_UNROLL; ++j) {
        long g = base + (long)j * CP_THREADS + tid;
        __builtin_amdgcn_global_store_async_from_lds_b128(
            GLOBAL_V4(dst + g), LDS_V4(&lbuf[j * CP_THREADS + tid]), 0, 0);
      }
      wait_async0();  // LDS reusable next iteration
#else
      u32x4 v[CP_UNROLL];
      #pragma unroll
      for (int j = 0; j < CP_UNROLL; ++j)
        v[j] = __builtin_nontemporal_load(src + base + (long)j * CP_THREADS + tid);
      #pragma unroll
      for (int j = 0; j < CP_UNROLL; ++j)
        __builtin_nontemporal_store(v[j], dst + base + (long)j * CP_THREADS + tid);
#endif
    } else {
      for (long g = base + tid; g < n16; g += CP_THREADS) dst[g] = src[g];
    }
  }
}

// ---------------------------------------------------------------------------
// Fixup: out row (l, r, tgt) = original input row (l, r, src), per cache.
// One B128 vector per thread; total = 2 * L * P * ROW_VECS transfers.
// ---------------------------------------------------------------------------
__global__ __launch_bounds__(256)
void fixup_kernel(const _Float16* __restrict__ K, const _Float16* __restrict__ V,
                  const int* __restrict__ req, const int* __restrict__ srcp,
                  const int* __restrict__ tgtp, _Float16* __restrict__ out,
                  long cache_elems, int P) {
  long gid = (long)blockIdx.x * blockDim.x + threadIdx.x;
  long total = 2L * L_DIM * (long)P * ROW_VECS;
  if (gid >= total) return;

  int v = (int)(gid % ROW_VECS);
  long t1 = gid / ROW_VECS;
  int p = (int)(t1 % P);
  long t2 = t1 / P;
  int l = (int)(t2 % L_DIM);
  int cache = (int)(t2 / L_DIM);  // 0 = K, 1 = V

  long r  = req[p];
  long s  = srcp[p];
  long tg = tgtp[p];

  const _Float16* src_base = cache ? V : K;
  long row_src = (((long)l * R_DIM + r) * T_DIM + s)  * ROW_ELEMS;
  long row_dst = (((long)l * R_DIM + r) * T_DIM + tg) * ROW_ELEMS;

  const u32x4* sp = (const u32x4*)(src_base + row_src);
  u32x4*       dp = (u32x4*)(out + (long)cache * cache_elems + row_dst);
  dp[v] = sp[v];
}

// ---------------------------------------------------------------------------
// Launch
// ---------------------------------------------------------------------------
extern "C" void kernel_launch(void* const* d_in, const int* in_sizes, int n_in,
                              void* d_out, int out_size, void* d_ws, size_t ws_size,
                              hipStream_t stream) {
  (void)n_in; (void)out_size; (void)d_ws; (void)ws_size;

  const _Float16* K    = (const _Float16*)d_in[0];
  const _Float16* V    = (const _Float16*)d_in[1];
  const int*      req  = (const int*)d_in[2];
  const int*      srcp = (const int*)d_in[3];
  const int*      tgtp = (const int*)d_in[4];
  _Float16*       out  = (_Float16*)d_out;

  const long nK = (long)in_sizes[0];  // L*R*T*H*D
  const long nV = (long)in_sizes[1];
  const int  P  = in_sizes[2];        // R*NUM_ACCEPT triples

  // Phase 1: stream both caches into the output (K first, then V).
  {
    long n16 = nK / 8;
    long tiles = (n16 + CP_TILE - 1) / CP_TILE;
    int grid = (int)(tiles < 8192 ? tiles : 8192);
    bulk_copy_kernel<<<grid, CP_THREADS, 0, stream>>>(
        (const u32x4*)K, (u32x4*)out, n16);
  }
  {
    long n16 = nV / 8;
    long tiles = (n16 + CP_TILE - 1) / CP_TILE;
    int grid = (int)(tiles < 8192 ? tiles : 8192);
    bulk_copy_kernel<<<grid, CP_THREADS, 0, stream>>>(
        (const u32x4*)V, (u32x4*)(out + nK), n16);
  }

  // Phase 2: overwrite the accepted rows from the ORIGINAL inputs.
  {
    long total = 2L * L_DIM * (long)P * ROW_VECS;
    int grid = (int)((total + 255) / 256);
    fixup_kernel<<<grid, 256, 0, stream>>>(K, V, req, srcp, tgtp, out, nK, P);
  }
}